// DiffraxSimulationWrapper_73177652789556
// MI455X (gfx1250) — compile-verified
//
#include <hip/hip_runtime.h>
#include <hip/hip_bf16.h>
#include <stdint.h>

typedef __attribute__((ext_vector_type(2))) float v2f;
typedef __attribute__((ext_vector_type(8))) float v8f;

#define BLOCK 128
#define MAX_STEPS 512

// ---------------- JAX-compatible threefry2x32 (20 rounds) ----------------
__device__ __forceinline__ void threefry2x32(uint32_t k0, uint32_t k1,
                                             uint32_t x0, uint32_t x1,
                                             uint32_t& o0, uint32_t& o1) {
  const uint32_t ks2 = k0 ^ k1 ^ 0x1BD11BDAu;
  x0 += k0; x1 += k1;
#define TF_R(d) { x0 += x1; x1 = (x1 << (d)) | (x1 >> (32 - (d))); x1 ^= x0; }
  TF_R(13) TF_R(15) TF_R(26) TF_R(6)  x0 += k1;  x1 += ks2 + 1u;
  TF_R(17) TF_R(29) TF_R(16) TF_R(24) x0 += ks2; x1 += k0 + 2u;
  TF_R(13) TF_R(15) TF_R(26) TF_R(6)  x0 += k0;  x1 += k1 + 3u;
  TF_R(17) TF_R(29) TF_R(16) TF_R(24) x0 += k1;  x1 += ks2 + 4u;
  TF_R(13) TF_R(15) TF_R(26) TF_R(6)  x0 += ks2; x1 += k0 + 5u;
#undef TF_R
  o0 = x0; o1 = x1;
}

// ---------------- XLA f32 erfinv (Giles) ----------------
__device__ __forceinline__ float erfinv32(float x) {
  float w = -log1pf(-x * x);
  float p;
  if (w < 5.0f) {
    w = w - 2.5f;
    p = 2.81022636e-08f;
    p = fmaf(p, w, 3.43273939e-07f);
    p = fmaf(p, w, -3.5233877e-06f);
    p = fmaf(p, w, -4.39150654e-06f);
    p = fmaf(p, w, 0.00021858087f);
    p = fmaf(p, w, -0.00125372503f);
    p = fmaf(p, w, -0.00417768164f);
    p = fmaf(p, w, 0.246640727f);
    p = fmaf(p, w, 1.50140941f);
  } else {
    w = sqrtf(w) - 3.0f;
    p = -0.000200214257f;
    p = fmaf(p, w, 0.000100950558f);
    p = fmaf(p, w, 0.00134934322f);
    p = fmaf(p, w, -0.00367342844f);
    p = fmaf(p, w, 0.00573950773f);
    p = fmaf(p, w, -0.0076224613f);
    p = fmaf(p, w, 0.00943887047f);
    p = fmaf(p, w, 1.00167406f);
    p = fmaf(p, w, 2.83297682f);
  }
  return p * x;
}

// bits -> uniform in [nextafter(-1,0), 1) -> standard normal (JAX semantics)
__device__ __forceinline__ float bits_to_normal(uint32_t bits) {
  float f = __uint_as_float((bits >> 9) | 0x3f800000u);  // [1,2)
  const float lo = -0.99999994f;                          // nextafterf(-1,0)
  float u = fmaf(f - 1.0f, 1.0f - lo, lo);
  u = fmaxf(u, lo);
  return 1.41421356237f * erfinv32(u);
}

__device__ __forceinline__ float clip01(float v) { return fminf(fmaxf(v, 0.0f), 1.0f); }

// trapezoidal PGSE envelope at time t (rise = dt)
__device__ __forceinline__ float envf(float t, float inv_dt, float delta, float Delta) {
  float p0 = clip01(t * inv_dt) - clip01((t - delta) * inv_dt);
  float p1 = clip01((t - Delta) * inv_dt) - clip01((t - Delta - delta) * inv_dt);
  return p0 - p1;
}

// ---------------- main simulation + phase-GEMM kernel ----------------
__global__ __launch_bounds__(BLOCK) void pgse_sim_kernel(
    const float* __restrict__ G_amps, const float* __restrict__ grads,
    const float* __restrict__ pDlong, const float* __restrict__ pDtrans,
    const int* __restrict__ p_delta_us, const int* __restrict__ p_Delta_us,
    const int* __restrict__ p_dt_us, const int* __restrict__ pN,
    const int* __restrict__ p_seed, float* __restrict__ partial, int M) {
  // per-step (k0, k1, weight, pad): one broadcast ds_load_b128 per RNG step
  __shared__ uint4 sKW[MAX_STEPS];
  // per-particle-slot (Sx, Sy, Sz, 0): one ds_load_b64 per WMMA B-fragment lane
  __shared__ float sS4[2 * BLOCK][4];

  const int tid = threadIdx.x;
  const int delta_us = p_delta_us[0];
  const int Delta_us = p_Delta_us[0];
  const int dt_us    = p_dt_us[0];
  const float dtf    = (float)dt_us * 1e-6f;
  const float inv_dt = 1.0f / dtf;
  const float deltaf = (float)delta_us * 1e-6f;
  const float Deltaf = (float)Delta_us * 1e-6f;
  const int total_us = delta_us + Delta_us + dt_us;
  const int T        = (total_us + dt_us - 1) / dt_us;   // == len(times)
  int steps          = T - 1;                             // number of noise draws
  if (steps > MAX_STEPS) steps = MAX_STEPS;
  const uint32_t seed_lo = (uint32_t)p_seed[0];

  // ---- stage 1: per-step folded keys + suffix-sum envelope weights ----
  for (int j = tid; j < steps; j += BLOCK) {
    uint32_t o0, o1;
    threefry2x32(0u, seed_lo, 0u, (uint32_t)j, o0, o1);  // fold_in(key, j)
    float wsum = 0.0f;
    for (int tt = j + 1; tt < T; ++tt)
      wsum += envf((float)tt * dtf, inv_dt, deltaf, Deltaf);
    sKW[j] = make_uint4(o0, o1, __float_as_uint(wsum), 0u);
  }
  __syncthreads();

  const int P = pN[0] >> 1;                 // particle pairs (threefry split point)
  const int totalChunks = (P + BLOCK - 1) / BLOCK;
  const uint32_t c1off = 3u * (uint32_t)P;  // counter offset of pair partner
  const float GDT  = 2.67513e8f * dtf;      // GAMMA * dt
  const float sigT = sqrtf(2.0f * dtf * pDtrans[0]);
  const float sigL = sqrtf(2.0f * dtf * pDlong[0]);

  // ---- constant A-tile per wave: 16 measurements x (3 axes + zero pad) ----
  const int lane = tid & 31, wv = tid >> 5;
  const int h = lane >> 4, jl = lane & 15;  // half-wave, row/col within tile
  const int mA = wv * 16 + jl;
  v2f a; a.x = 0.0f; a.y = 0.0f;
  if (mA < M) {
    float gG = GDT * G_amps[mA];
    if (h == 0) { a.x = gG * grads[3 * mA + 0]; a.y = gG * grads[3 * mA + 1]; }
    else        { a.x = gG * grads[3 * mA + 2]; a.y = 0.0f; }
  }

  float accc[8], accs[8];
#pragma unroll
  for (int r = 0; r < 8; ++r) { accc[r] = 0.0f; accs[r] = 0.0f; }

  // ---- grid-stride over particle chunks (deterministic, ws-capacity safe) ----
  for (int chunk = blockIdx.x; chunk < totalChunks; chunk += gridDim.x) {
    const int t = chunk * BLOCK + tid;
    float S0x = 0.f, S0y = 0.f, S0z = 0.f, S1x = 0.f, S1y = 0.f, S1z = 0.f;
    if (t < P) {
      const uint32_t cb = 3u * (uint32_t)t;
      for (int j = 0; j < steps; ++j) {
        const uint4 kw = sKW[j];
        const uint32_t k0 = kw.x, k1 = kw.y;
        const float wj = __uint_as_float(kw.z);
        uint32_t o0, o1;
        threefry2x32(k0, k1, cb + 0u, cb + 0u + c1off, o0, o1);
        S0x = fmaf(wj, bits_to_normal(o0), S0x);
        S1x = fmaf(wj, bits_to_normal(o1), S1x);
        threefry2x32(k0, k1, cb + 1u, cb + 1u + c1off, o0, o1);
        S0y = fmaf(wj, bits_to_normal(o0), S0y);
        S1y = fmaf(wj, bits_to_normal(o1), S1y);
        threefry2x32(k0, k1, cb + 2u, cb + 2u + c1off, o0, o1);
        S0z = fmaf(wj, bits_to_normal(o0), S0z);
        S1z = fmaf(wj, bits_to_normal(o1), S1z);
      }
    }
    __syncthreads();  // protect LDS S from previous chunk's phase stage
    *(float4*)&sS4[tid][0]         = make_float4(S0x * sigT, S0y * sigT, S0z * sigL, 0.0f);
    *(float4*)&sS4[BLOCK + tid][0] = make_float4(S1x * sigT, S1y * sigT, S1z * sigL, 0.0f);
    __syncthreads();

    // ---- phase stage: D(16x16) = A(16x4) x B(4x16) via v_wmma_f32_16x16x4_f32
    // wave wv covers measurements [16wv,16wv+16); loop covers all 256 particles.
    // B fragment: half h loads (Sx,Sy) or (Sz,0) -> single branchless ds_load_b64.
#pragma unroll 4
    for (int g = 0; g < 16; ++g) {
      const int n = g * 16 + jl;  // particle slot this lane's column reads
      const v2f bb = *(const v2f*)&sS4[n][h << 1];
      v8f cz = {};
      v8f dm = __builtin_amdgcn_wmma_f32_16x16x4_f32(
          false, a, false, bb, (short)0, cz, false, false);
      const float mk = ((chunk * BLOCK + (n & (BLOCK - 1))) < P) ? 1.0f : 0.0f;
#pragma unroll
      for (int r = 0; r < 8; ++r) {
        float sv, cv;
        __sincosf(dm[r], &sv, &cv);
        accc[r] = fmaf(mk, cv, accc[r]);
        accs[r] = fmaf(mk, sv, accs[r]);
      }
    }
  }

  // ---- reduce the 16 columns within each half-wave, write block partials ----
#pragma unroll
  for (int off = 1; off < 16; off <<= 1) {
#pragma unroll
    for (int r = 0; r < 8; ++r) {
      accc[r] += __shfl_xor(accc[r], off, 32);
      accs[r] += __shfl_xor(accs[r], off, 32);
    }
  }
  if (jl == 0) {
#pragma unroll
    for (int r = 0; r < 8; ++r) {
      const int m = wv * 16 + 8 * h + r;  // D-tile row -> measurement index
      partial[blockIdx.x * 128 + m]      = accc[r];
      partial[blockIdx.x * 128 + 64 + m] = accs[r];
    }
  }
}

// ---------------- deterministic cross-block reduction + magnitude ----------------
__global__ void pgse_finalize_kernel(const float* __restrict__ partial, int nb,
                                     const int* __restrict__ pN,
                                     float* __restrict__ out, int M) {
  const int m = blockIdx.x * blockDim.x + threadIdx.x;
  if (m >= M) return;
  float c = 0.0f, s = 0.0f;
  for (int i = 0; i < nb; ++i) {
    c += partial[i * 128 + m];
    s += partial[i * 128 + 64 + m];
  }
  const float invN = 1.0f / (float)pN[0];
  c *= invN; s *= invN;
  out[m] = sqrtf(c * c + s * s);
}

extern "C" void kernel_launch(void* const* d_in, const int* in_sizes, int n_in,
                              void* d_out, int out_size, void* d_ws, size_t ws_size,
                              hipStream_t stream) {
  const float* G_amps   = (const float*)d_in[0];
  const float* grads    = (const float*)d_in[1];
  const float* D_long   = (const float*)d_in[2];
  const float* D_trans  = (const float*)d_in[3];
  const int*   delta_us = (const int*)d_in[4];
  const int*   Delta_us = (const int*)d_in[5];
  const int*   dt_us    = (const int*)d_in[6];
  const int*   Np       = (const int*)d_in[7];
  const int*   seed     = (const int*)d_in[8];
  float* out = (float*)d_out;
  float* partial = (float*)d_ws;

  const int M = in_sizes[0];  // number of measurements (64)
  // Host-side parallelism hint from reference setup (N=50000 -> 25000 pairs);
  // actual loop bounds are device-read, grid-stride covers any excess.
  int nb = (25000 + BLOCK - 1) / BLOCK;  // 196 blocks
  const size_t cap = ws_size / (128 * sizeof(float));
  if (cap > 0 && (size_t)nb > cap) nb = (int)cap;
  if (nb < 1) nb = 1;

  pgse_sim_kernel<<<nb, BLOCK, 0, stream>>>(G_amps, grads, D_long, D_trans,
                                            delta_us, Delta_us, dt_us, Np, seed,
                                            partial, M);
  pgse_finalize_kernel<<<(M + 63) / 64, 64, 0, stream>>>(partial, nb, Np, out, M);
}